// TANDemoPredictor_84121229460056
// MI455X (gfx1250) — compile-verified
//
#include <hip/hip_runtime.h>
#include <math.h>

typedef __attribute__((ext_vector_type(2))) float v2f;
typedef __attribute__((ext_vector_type(8))) float v8f;

#define B_  4096
#define S_  200
#define D_  64
#define NK  5      // attention heads
#define NC  18     // total classes (2+4+4+2+6)

__device__ __forceinline__ v8f wmma_f32_16x16x4(v2f a, v2f b, v8f c) {
#if __has_builtin(__builtin_amdgcn_wmma_f32_16x16x4_f32)
  return __builtin_amdgcn_wmma_f32_16x16x4_f32(false, a, false, b, (short)0, c,
                                               false, false);
#else
  #pragma unroll
  for (int j = 0; j < 8; ++j) c[j] += a.x * b.x + a.y * b.y;
  return c;
#endif
}

__device__ __forceinline__ void wait_asynccnt0() {
#if __has_builtin(__builtin_amdgcn_s_wait_asynccnt)
  __builtin_amdgcn_s_wait_asynccnt(0);
#else
  asm volatile("s_wait_asynccnt 0x0" ::: "memory");
#endif
}

__global__ void tan_zero_loss(float* out) { out[B_ * NC] = 0.0f; }

__global__ __launch_bounds__(256) void tan_fused_kernel(
    const int*   __restrict__ x,      // [B,S]
    const float* __restrict__ y,      // [B,18]
    const float* __restrict__ emb,    // [V,64]
    const float* __restrict__ att_w,  // [5,64]
    const float* __restrict__ att_b,  // [5]
    const float* __restrict__ W0,     // [2,64]
    const float* __restrict__ W1,     // [4,64]
    const float* __restrict__ W2,     // [4,64]
    const float* __restrict__ W3,     // [2,64]
    const float* __restrict__ W4,     // [6,64]
    float* __restrict__ out)          // [B*18 logits][1 loss]
{
  __shared__ __align__(16) float s_embed[S_ * D_];      // 51.2 KB
  __shared__ float s_att[S_ * NK + 4];                  // att_u, then score
  __shared__ __align__(16) float s_attw[NK * D_];
  __shared__ float s_attb[8];
  __shared__ float s_user[NK * D_];                     // user_rep [5,64]
  __shared__ float s_wuser[24];                         // 18 raw logits
  __shared__ float s_loss[8];
  __shared__ int   s_xi[S_];

  const int tid  = threadIdx.x;
  const int lane = tid & 31;
  const int wid  = tid >> 5;
  const int b    = blockIdx.x;

  // ---- stage small weights + the 200 indices for this batch row ----
  for (int i = tid; i < NK * D_; i += 256) s_attw[i] = att_w[i];
  if (tid < NK) s_attb[tid] = att_b[tid];
  if (tid < S_) s_xi[tid] = x[b * S_ + tid];
  __syncthreads();

  // ---- gather embeddings straight into LDS via async global->LDS B128 ----
  // 200 rows x 16 float4; ASYNCcnt-tracked, no VGPR round trip.
  {
    const unsigned lds_base =
        (unsigned)(unsigned long long)(uintptr_t)(&s_embed[0]);
    for (int i = tid; i < S_ * (D_ / 4); i += 256) {
      const int s = i >> 4;
      const int q = i & 15;
      const int v = s_xi[s];
      const float* gp = emb + (long)v * D_ + q * 4;
      const unsigned lds_addr = lds_base + (unsigned)(i * 16);
      asm volatile("global_load_async_to_lds_b128 %0, %1, off"
                   :
                   : "v"(lds_addr), "v"(gp)
                   : "memory");
    }
    wait_asynccnt0();
  }
  __syncthreads();

  // ---- att_u = tanh(embed @ att_w^T + b) via fp32 WMMA 16x16x4 ----
  // 13 S-tiles of 16 rows; K = 64 in steps of 4. Wave-uniform tiling.
  {
    const int n    = lane & 15;                 // output column (head)
    const int nb   = (n < NK) ? n : (NK - 1);   // clamp (cols >=5 discarded)
    const int koff = (lane & 16) ? 2 : 0;
    for (int tile = wid; tile < 13; tile += 8) {
      const int row  = tile * 16 + (lane & 15);
      const int rowc = (row < S_) ? row : (S_ - 1);  // clamp (rows discarded)
      v8f c = {0.f, 0.f, 0.f, 0.f, 0.f, 0.f, 0.f, 0.f};
      #pragma unroll
      for (int kb = 0; kb < D_; kb += 4) {
        v2f a, bm;
        a.x  = s_embed[rowc * D_ + kb + koff];
        a.y  = s_embed[rowc * D_ + kb + koff + 1];
        bm.x = s_attw[nb * D_ + kb + koff];
        bm.y = s_attw[nb * D_ + kb + koff + 1];
        c = wmma_f32_16x16x4(a, bm, c);
      }
      const int rbase = tile * 16 + ((lane & 16) ? 8 : 0);
      if (n < NK) {
        #pragma unroll
        for (int j = 0; j < 8; ++j) {
          const int r = rbase + j;
          if (r < S_) s_att[r * NK + n] = tanhf(c[j] + s_attb[n]);
        }
      }
    }
  }
  __syncthreads();

  // ---- softmax over S per head: wave k handles head k (wave32 shfl) ----
  if (wid < NK) {
    const int k = wid;
    float m = -3.402823e38f;
    for (int s = lane; s < S_; s += 32) m = fmaxf(m, s_att[s * NK + k]);
    #pragma unroll
    for (int off = 16; off >= 1; off >>= 1) m = fmaxf(m, __shfl_xor(m, off, 32));
    float sum = 0.f;
    for (int s = lane; s < S_; s += 32) sum += expf(s_att[s * NK + k] - m);
    #pragma unroll
    for (int off = 16; off >= 1; off >>= 1) sum += __shfl_xor(sum, off, 32);
    const float inv = 1.0f / sum;
    for (int s = lane; s < S_; s += 32)
      s_att[s * NK + k] = expf(s_att[s * NK + k] - m) * inv;   // score
  }
  __syncthreads();

  // ---- user_rep = score^T @ embed  (M=5->16, N=64, K=200) via WMMA ----
  // waves 0..3 each own one 16-wide slice of D.
  if (wid < 4) {
    const int dt   = wid * 16;
    const int n    = lane & 15;                 // D column within slice
    const int mc   = ((lane & 15) < NK) ? (lane & 15) : (NK - 1); // clamp M
    const int koff = (lane & 16) ? 2 : 0;
    v8f c = {0.f, 0.f, 0.f, 0.f, 0.f, 0.f, 0.f, 0.f};
    for (int s0 = 0; s0 < S_; s0 += 4) {
      v2f a, bm;
      a.x  = s_att[(s0 + koff) * NK + mc];
      a.y  = s_att[(s0 + koff + 1) * NK + mc];
      bm.x = s_embed[(s0 + koff) * D_ + dt + n];
      bm.y = s_embed[(s0 + koff + 1) * D_ + dt + n];
      c = wmma_f32_16x16x4(a, bm, c);
    }
    const int rbase = (lane & 16) ? 8 : 0;
    #pragma unroll
    for (int j = 0; j < 8; ++j) {
      const int k = rbase + j;
      if (k < NK) s_user[k * D_ + dt + n] = c[j];
    }
  }
  __syncthreads();

  // ---- head projections: 18 raw logits ----
  if (tid < NC) {
    const int cum[6] = {0, 2, 6, 10, 12, 18};
    int head = 0;
    #pragma unroll
    for (int i = 0; i < 5; ++i) if (tid >= cum[i + 1]) head = i + 1;
    const int cidx = tid - cum[head];
    const float* Wp;
    switch (head) {
      case 0: Wp = W0; break;
      case 1: Wp = W1; break;
      case 2: Wp = W2; break;
      case 3: Wp = W3; break;
      default: Wp = W4; break;
    }
    float acc = 0.f;
    #pragma unroll 8
    for (int d = 0; d < D_; ++d) acc += s_user[head * D_ + d] * Wp[cidx * D_ + d];
    s_wuser[tid] = acc;
  }
  __syncthreads();

  // ---- per-head softmax / log-softmax + loss (threads 0..4) ----
  if (tid < NK) {
    const int cum[6] = {0, 2, 6, 10, 12, 18};
    const int len[5] = {2, 4, 4, 2, 6};
    const int off = cum[tid], L = len[tid];
    float m = -3.402823e38f;
    for (int c = 0; c < 6; ++c) if (c < L) m = fmaxf(m, s_wuser[off + c]);
    float sum = 0.f;
    for (int c = 0; c < 6; ++c) if (c < L) sum += expf(s_wuser[off + c] - m);
    const float lse = m + logf(sum);
    const float inv = 1.0f / sum;
    float lloss = 0.f;
    for (int c = 0; c < 6; ++c) {
      if (c < L) {
        const float z = s_wuser[off + c];
        out[b * NC + off + c] = expf(z - m) * inv;          // softmax logits
        lloss += (lse - z) * y[b * NC + off + c];           // -logp * y
      }
    }
    s_loss[tid] = lloss;
  }
  __syncthreads();
  if (tid == 0) {
    const float t = s_loss[0] + s_loss[1] + s_loss[2] + s_loss[3] + s_loss[4];
    atomicAdd(out + B_ * NC, t * (1.0f / (float)B_));
  }
}

extern "C" void kernel_launch(void* const* d_in, const int* in_sizes, int n_in,
                              void* d_out, int out_size, void* d_ws, size_t ws_size,
                              hipStream_t stream) {
  (void)in_sizes; (void)n_in; (void)out_size; (void)d_ws; (void)ws_size;
  const int*   x     = (const int*)  d_in[0];
  const float* y     = (const float*)d_in[1];
  const float* emb   = (const float*)d_in[2];
  const float* att_w = (const float*)d_in[3];
  const float* att_b = (const float*)d_in[4];
  const float* W0    = (const float*)d_in[5];
  const float* W1    = (const float*)d_in[6];
  const float* W2    = (const float*)d_in[7];
  const float* W3    = (const float*)d_in[8];
  const float* W4    = (const float*)d_in[9];
  float* out = (float*)d_out;

  tan_zero_loss<<<1, 1, 0, stream>>>(out);
  tan_fused_kernel<<<B_, 256, 0, stream>>>(x, y, emb, att_w, att_b,
                                           W0, W1, W2, W3, W4, out);
}